// GINConvolution_35966056137207
// MI455X (gfx1250) — compile-verified
//
#include <hip/hip_runtime.h>

typedef __attribute__((ext_vector_type(2))) float v2f;
typedef __attribute__((ext_vector_type(8))) float v8f;

#define N_NODES 100000
#define N_EDGES 1600000
#define D_IN 128
#define D_HID 256
#define D_OUT 128

#define LDH  132   // padded LDS stride for h tile   (132 % 64 = 4 -> conflict-free)
#define LDH1 260   // padded LDS stride for h1 tile  (260 % 64 = 4 -> conflict-free)

// ---------------------------------------------------------------------------
// Phase 0: h = (1 + eps) * x          (vectorized float4 init of workspace)
// ---------------------------------------------------------------------------
__global__ void gin_init_h(const float* __restrict__ x,
                           const float* __restrict__ eps,
                           float* __restrict__ h, int n4) {
  int i = blockIdx.x * blockDim.x + threadIdx.x;
  if (i < n4) {
    float s = 1.0f + eps[0];
    float4 v = ((const float4*)x)[i];
    v.x *= s; v.y *= s; v.z *= s; v.w *= s;
    ((float4*)h)[i] = v;
  }
}

// ---------------------------------------------------------------------------
// Phase 1: scatter  h[dst] += val * x[src]
// One wave32 per edge: 32 lanes x float4 = coalesced 512B gather of x[src],
// then 4 global_atomic_add_f32 per lane (atomics resolve in the 192MB L2;
// h is 51 MB and fully L2-resident).
// ---------------------------------------------------------------------------
__global__ void gin_scatter(const float* __restrict__ x,
                            const float* __restrict__ vals,
                            const int* __restrict__ src,
                            const int* __restrict__ dst,
                            float* __restrict__ h) {
  int wave = threadIdx.x >> 5;
  int lane = threadIdx.x & 31;
  int e = blockIdx.x * 8 + wave;
  if (e >= N_EDGES) return;
  int   s = src[e];
  int   d = dst[e];
  float v = vals[e];
  float4 xv = ((const float4*)(x + (size_t)s * D_IN))[lane];
  float* hd = h + (size_t)d * D_IN + lane * 4;
  atomicAdd(hd + 0, v * xv.x);
  atomicAdd(hd + 1, v * xv.y);
  atomicAdd(hd + 2, v * xv.z);
  atomicAdd(hd + 3, v * xv.w);
}

// ---------------------------------------------------------------------------
// Phase 2: fused MLP  out = relu(h @ W1 + b1) @ W2 + b2
// wave32 WMMA f32 16x16x4. One wave owns a 32-row tile (two 16-row
// M-subtiles) so every B fragment (W1/W2) is loaded once and fed to two
// WMMAs -> 1 global load per WMMA instead of 2, and half the L2 weight
// traffic. h1 tile lives entirely in LDS (never touches HBM).
// 1 wave / block, 3125 blocks (100000 = 3125 * 32, no tail).
// ---------------------------------------------------------------------------
__global__ void __launch_bounds__(32)
gin_mlp(const float* __restrict__ h,
        const float* __restrict__ W1, const float* __restrict__ b1,
        const float* __restrict__ W2, const float* __restrict__ b2,
        float* __restrict__ out) {
  __shared__ float sh [32][LDH];    // h tile  32x128
  __shared__ float sh1[32][LDH1];   // h1 tile 32x256

  int lane = threadIdx.x & 31;
  int m0   = blockIdx.x * 32;

  // ---- stage 32x128 h tile into LDS (float4, coalesced) ----
  for (int i = lane; i < 32 * (D_IN / 4); i += 32) {
    int row = i >> 5;          // D_IN/4 = 32 float4 per row
    int c4  = i & 31;
    float4 v = ((const float4*)(h + (size_t)(m0 + row) * D_IN))[c4];
    sh[row][c4 * 4 + 0] = v.x;
    sh[row][c4 * 4 + 1] = v.y;
    sh[row][c4 * 4 + 2] = v.z;
    sh[row][c4 * 4 + 3] = v.w;
  }

  // WMMA f32 16x16x4 fragment geometry (ISA 7.12.2):
  //   A: lanes 0-15 -> M=lane, K=k0..k0+1 ; lanes 16-31 -> M=lane-16, K=k0+2..k0+3
  //   B: mirrored (N=lane&15, same K split)
  //   C: vgpr r -> M = r + 8*(lane>>4), N = lane&15
  int half = lane >> 4;        // 0 or 1
  int rA   = lane & 15;        // A row / B col within 16x16 tile
  int kOff = half * 2;
  int mB   = half * 8;

  // ---- stage 1: h1 = relu(h @ W1 + b1), kept in LDS ----
  for (int nt = 0; nt < D_HID / 16; ++nt) {
    int n0 = nt * 16;
    v8f acc0 = {};
    v8f acc1 = {};
#pragma unroll 8
    for (int kb = 0; kb < D_IN / 4; ++kb) {
      int k0 = kb * 4 + kOff;
      v2f a0 = *(const v2f*)&sh[rA][k0];        // M-subtile 0: rows 0..15
      v2f a1 = *(const v2f*)&sh[16 + rA][k0];   // M-subtile 1: rows 16..31
      const float* wp = W1 + (size_t)k0 * D_HID + n0 + rA;
      v2f b; b.x = wp[0]; b.y = wp[D_HID];
      acc0 = __builtin_amdgcn_wmma_f32_16x16x4_f32(
          false, a0, false, b, (short)0, acc0, false, false);
      acc1 = __builtin_amdgcn_wmma_f32_16x16x4_f32(
          false, a1, false, b, (short)0, acc1, false, false);
    }
    float bias = b1[n0 + rA];
#pragma unroll
    for (int r = 0; r < 8; ++r) {
      float v0 = acc0[r] + bias;
      float v1 = acc1[r] + bias;
      sh1[mB + r][n0 + rA]      = v0 > 0.0f ? v0 : 0.0f;
      sh1[16 + mB + r][n0 + rA] = v1 > 0.0f ? v1 : 0.0f;
    }
  }

  // ---- stage 2: out = h1 @ W2 + b2 ----
  for (int nt = 0; nt < D_OUT / 16; ++nt) {
    int n0 = nt * 16;
    v8f acc0 = {};
    v8f acc1 = {};
#pragma unroll 8
    for (int kb = 0; kb < D_HID / 4; ++kb) {
      int k0 = kb * 4 + kOff;
      v2f a0 = *(const v2f*)&sh1[rA][k0];
      v2f a1 = *(const v2f*)&sh1[16 + rA][k0];
      const float* wp = W2 + (size_t)k0 * D_OUT + n0 + rA;
      v2f b; b.x = wp[0]; b.y = wp[D_OUT];
      acc0 = __builtin_amdgcn_wmma_f32_16x16x4_f32(
          false, a0, false, b, (short)0, acc0, false, false);
      acc1 = __builtin_amdgcn_wmma_f32_16x16x4_f32(
          false, a1, false, b, (short)0, acc1, false, false);
    }
    float bias = b2[n0 + rA];
#pragma unroll
    for (int r = 0; r < 8; ++r) {
      out[(size_t)(m0 + mB + r) * D_OUT + n0 + rA]      = acc0[r] + bias;
      out[(size_t)(m0 + 16 + mB + r) * D_OUT + n0 + rA] = acc1[r] + bias;
    }
  }
}

// ---------------------------------------------------------------------------
extern "C" void kernel_launch(void* const* d_in, const int* in_sizes, int n_in,
                              void* d_out, int out_size, void* d_ws, size_t ws_size,
                              hipStream_t stream) {
  (void)in_sizes; (void)n_in; (void)out_size; (void)ws_size;
  const float* x    = (const float*)d_in[0];
  const float* ev   = (const float*)d_in[1];
  const float* W1   = (const float*)d_in[2];
  const float* b1   = (const float*)d_in[3];
  const float* W2   = (const float*)d_in[4];
  const float* b2   = (const float*)d_in[5];
  const float* eps  = (const float*)d_in[6];
  const int*   esrc = (const int*)d_in[7];
  const int*   edst = (const int*)d_in[8];
  float* out = (float*)d_out;
  float* h   = (float*)d_ws;                 // N_NODES * D_IN floats (51.2 MB)

  int n4 = N_NODES * D_IN / 4;
  gin_init_h<<<(n4 + 255) / 256, 256, 0, stream>>>(x, eps, h, n4);
  gin_scatter<<<(N_EDGES + 7) / 8, 256, 0, stream>>>(x, ev, esrc, edst, h);
  gin_mlp<<<N_NODES / 32, 32, 0, stream>>>(h, W1, b1, W2, b2, out);
}